// HopfLayer_56556129354060
// MI455X (gfx1250) — compile-verified
//
#include <hip/hip_runtime.h>
#include <cstdint>

#define T_STEPS 4096
#define UNITS   256
#define BATCH   64
#define UB      64      // units per block (2 wave32s)
#define TC      8       // time steps per chunk (16 async ops per chunk)
#define NBUF    4       // 4-deep async pipeline: lead = 3 chunks = 24 steps
#define DT      0.001f
#define BETA    0.01f
#define INSCALE 0.1f

// Async global->LDS copy, GVS mode: uniform 64-bit SGPR base + per-lane 32-bit byte offset.
__device__ __forceinline__ void async_ld(uint32_t lds_off, uint32_t gbyte_off, const float* base) {
    asm volatile("global_load_async_to_lds_b32 %0, %1, %2"
                 :: "v"(lds_off), "v"(gbyte_off), "s"(base)
                 : "memory");
}

__global__ void __launch_bounds__(UB)
hopf_rollout_kernel(const float* __restrict__ x_r,
                    const float* __restrict__ x_i,
                    const float* __restrict__ omegas,
                    float* __restrict__ out) {
    // 4-deep staging tiles: 4 * 8 * 64 * 4B * 2 arrays = 16 KB LDS
    __shared__ float xr_s[NBUF][TC][UB];
    __shared__ float xi_s[NBUF][TC][UB];

    const int ul    = threadIdx.x;
    const int blk   = blockIdx.x;
    const int b     = blk >> 2;             // UNITS/UB = 4 blocks per batch row
    const int ubase = (blk & 3) * UB;
    const int u     = ubase + ul;

    const float omega = omegas[u];

    // Byte offset of element (b, t=0, u); max ~268 MB fits in u32.
    const uint32_t goff = (((unsigned)b * T_STEPS) * UNITS + (unsigned)u) * 4u;

    const uint32_t lr0 = (uint32_t)(uintptr_t)(&xr_s[0][0][ul]);
    const uint32_t li0 = (uint32_t)(uintptr_t)(&xi_s[0][0][ul]);
    const uint32_t BUFSTRIDE = TC * UB * 4;        // 2048 B between buffers
    const uint32_t CHUNK_GB  = TC * UNITS * 4;     // 8192 B of global per chunk

    const int NCH = T_STEPS / TC;                  // 512 chunks

    // Helper to issue one chunk's async loads (16 instructions).
    auto issue_chunk = [&](int k) {
        const uint32_t boff = (uint32_t)(k & (NBUF - 1)) * BUFSTRIDE;
        uint32_t g  = goff + (uint32_t)k * CHUNK_GB;
        uint32_t lr = lr0 + boff;
        uint32_t li = li0 + boff;
#pragma unroll
        for (int tt = 0; tt < TC; ++tt) {
            async_ld(lr, g, x_r);
            async_ld(li, g, x_i);
            g  += UNITS * 4;
            lr += UB * 4;
            li += UB * 4;
        }
    };

    // ---- prologue: stage chunks 0..2 (48 async ops in flight) ----
    issue_chunk(0);
    issue_chunk(1);
    issue_chunk(2);

    // Oscillator state; cos/sin of the current phase carried across steps
    // (output factors of step t == coupling terms of step t+1).
    float r = 1.0f, phi = 0.0f, c = 1.0f, s = 0.0f;

    const size_t half = (size_t)BATCH * T_STEPS * UNITS;
    float* pr = out + ((size_t)b * T_STEPS) * UNITS + (size_t)u;
    float* pi = pr + half;

    for (int k = 0; k < NCH; ++k) {
        if (k + 3 < NCH) {
            issue_chunk(k + 3);
            asm volatile("s_wait_asynccnt 0x30" ::: "memory"); // 3 chunks outstanding -> chunk k landed
        } else if (k + 2 < NCH) {
            asm volatile("s_wait_asynccnt 0x20" ::: "memory");
        } else if (k + 1 < NCH) {
            asm volatile("s_wait_asynccnt 0x10" ::: "memory");
        } else {
            asm volatile("s_wait_asynccnt 0x0" ::: "memory");
        }

        const int cbuf = k & (NBUF - 1);
#pragma unroll
        for (int tt = 0; tt < TC; ++tt) {
            const float xr = xr_s[cbuf][tt][ul];
            const float xi = xi_s[cbuf][tt][ul];

            const float in_r = INSCALE * xr * c;   // cos(phi_old)
            const float in_p = INSCALE * xi * s;   // sin(phi_old)

            r   = r + ((1.0f - BETA * r * r) * r + in_r) * DT;
            phi = phi + (omega - in_p) * DT;

            c = __cosf(phi);                        // v_cos_f32 path
            s = __sinf(phi);                        // v_sin_f32 path

            // 512 MB output stream > 192 MB L2: non-temporal streaming stores.
            __builtin_nontemporal_store(r * c, pr);
            __builtin_nontemporal_store(r * s, pi);
            pr += UNITS;
            pi += UNITS;
        }
    }
}

extern "C" void kernel_launch(void* const* d_in, const int* in_sizes, int n_in,
                              void* d_out, int out_size, void* d_ws, size_t ws_size,
                              hipStream_t stream) {
    const float* x_r    = (const float*)d_in[0];
    const float* x_i    = (const float*)d_in[1];
    const float* omegas = (const float*)d_in[2];
    float* out = (float*)d_out;

    dim3 grid(BATCH * (UNITS / UB));   // 256 blocks spread across WGPs
    dim3 block(UB);                    // 64 threads = 2 wave32
    hipLaunchKernelGGL(hopf_rollout_kernel, grid, block, 0, stream,
                       x_r, x_i, omegas, out);
}